// HUXPropagator_82884278878694
// MI455X (gfx1250) — compile-verified
//
#include <hip/hip_runtime.h>
#include <stdint.h>

#define T_PHI   4096
#define NT      512          // threads per block (16 wave32 waves)
#define E       8            // elements owned per thread
#define HALO    8            // halo depth = steps per exchange round
#define NRADIUS 512
#define NROUND  64           // 63 rounds * 8 steps + 1 round * 7 steps = 511

// Fast division: v_rcp_f32 + 1 Newton-Raphson + quotient correction (~0.5 ulp).
__device__ __forceinline__ float fdiv_fast(float a, float b) {
  float r = __builtin_amdgcn_rcpf(b);
  r = fmaf(r, fmaf(-b, r, 1.0f), r);      // NR refine reciprocal
  float q = a * r;
  q = fmaf(fmaf(-b, q, a), r, q);          // correct the quotient
  return q;
}

__global__ __launch_bounds__(NT, 2) void hux_propagate_kernel(
    const float* __restrict__ v_in,
    const float* __restrict__ omega,
    float* __restrict__ v_out) {
  __shared__ float xbuf[2][T_PHI];         // double-buffered exchange slab (32 KB)

  const int tid = threadIdx.x;
  const int b   = blockIdx.x;

  // Match reference constant construction: f64 math, then cast to f32.
  const float DR_KM = (float)(((1.0 - 0.1) / (double)(NRADIUS - 1)) * 149600000.0);
  const float DPHI  = (float)(6.283185307179586 / (double)T_PHI);
  const float coeff = DR_KM * omega[b] / DPHI;

  // ---- CDNA5 async global -> LDS: stage the whole row into xbuf[0] ----
  // xbuf is the only __shared__ object => LDS offset 0. Each lane moves 32B
  // as two b128 async transfers (GVS mode: saddr = row base, vaddr = offset).
  const float* rowp = v_in + (size_t)b * T_PHI;
  {
    unsigned off0 = (unsigned)tid * 32u;
    unsigned off1 = off0 + 16u;
    asm volatile("global_load_async_to_lds_b128 %0, %1, %2"
                 :: "v"(off0), "v"(off0), "s"(rowp) : "memory");
    asm volatile("global_load_async_to_lds_b128 %0, %1, %2"
                 :: "v"(off1), "v"(off1), "s"(rowp) : "memory");
    asm volatile("s_wait_asynccnt 0" ::: "memory");
  }
  __syncthreads();   // neighbors' async deposits visible to everyone

  const int slot  = tid * E;                       // my 8 elements
  const int nslot = ((tid + 1) & (NT - 1)) * E;    // circular right neighbor

  float w[E + HALO];
#pragma unroll
  for (int i = 0; i < E; ++i) w[i] = xbuf[0][slot + i];

  for (int r = 0; r < NROUND; ++r) {
    const int parity = r & 1;
    const int L = (r == NROUND - 1) ? 7 : 8;       // 63*8 + 7 = 511 steps

    if (r != 0) {
      // Publish current state; double buffering makes 1 barrier/round safe.
      float4* dst = (float4*)&xbuf[parity][slot];
      dst[0] = make_float4(w[0], w[1], w[2], w[3]);
      dst[1] = make_float4(w[4], w[5], w[6], w[7]);
      __syncthreads();
    }

    // Pull 8-element halo from circular neighbor.
    const float4* hsrc = (const float4*)&xbuf[parity][nslot];
    float4 h0 = hsrc[0], h1 = hsrc[1];
    w[E + 0] = h0.x; w[E + 1] = h0.y; w[E + 2] = h0.z; w[E + 3] = h0.w;
    w[E + 4] = h1.x; w[E + 5] = h1.y; w[E + 6] = h1.z; w[E + 7] = h1.w;

    // Trapezoidal local advance: sub-step s has valid window w[0 .. 15-s].
    // Left-to-right in-place update reads only old values (w[i+1] untouched).
#pragma unroll
    for (int s = 0; s < HALO; ++s) {
      if (s < L) {
#pragma unroll
        for (int i = 0; i < E + HALO - 1 - s; ++i) {
          float v  = w[i];
          float vn = w[i + 1];
          w[i] = fmaf(coeff, fdiv_fast(vn - v, v), v);
        }
      }
    }
  }

  // Flipped store: out[T-1-t] = v_final[t]. For my block of 8 this is two
  // reversed, 32B-aligned b128 stores (still contiguous across the wave).
  float* orow = v_out + (size_t)b * T_PHI;
  const int base = T_PHI - E - slot;
  float4* o = (float4*)&orow[base];
  o[0] = make_float4(w[7], w[6], w[5], w[4]);
  o[1] = make_float4(w[3], w[2], w[1], w[0]);
}

extern "C" void kernel_launch(void* const* d_in, const int* in_sizes, int n_in,
                              void* d_out, int out_size, void* d_ws, size_t ws_size,
                              hipStream_t stream) {
  const float* v_in  = (const float*)d_in[0];   // [B, T] f32
  const float* omega = (const float*)d_in[1];   // [B]    f32
  float* out = (float*)d_out;                   // [B, T] f32
  const int B = in_sizes[1];                    // 64
  hux_propagate_kernel<<<dim3(B), dim3(NT), 0, stream>>>(v_in, omega, out);
}